// MultiHeadAttention_58110907515489
// MI455X (gfx1250) — compile-verified
//
#include <hip/hip_runtime.h>
#include <math.h>

#define N_HEADS   6
#define HEAD_SIZE 16
#define N_EMBD    96
#define BATCH     2
#define SEQ       4096

typedef __attribute__((ext_vector_type(2))) float v2f;
typedef __attribute__((ext_vector_type(8))) float v8f;

// ---------------------------------------------------------------------------
// Kernel 1: QKV projection.
//   q[bh][t][d], k[bh][t][d]  (row-major per head, hs=16 contiguous)
//   vT[bh][d][t]              (transposed so PV A-operands are contiguous)
// ---------------------------------------------------------------------------
__global__ void qkv_proj_kernel(const float* __restrict__ x,
                                const float* __restrict__ Wq,
                                const float* __restrict__ Wk,
                                const float* __restrict__ Wv,
                                float* __restrict__ qo,
                                float* __restrict__ ko,
                                float* __restrict__ vT) {
  int idx = blockIdx.x * blockDim.x + threadIdx.x;   // B*H*T*16 threads
  int d   = idx & 15;
  int t   = (idx >> 4) & (SEQ - 1);
  int bh  = idx >> 16;                               // 4 + 12 bits
  int b   = bh / N_HEADS;
  int h   = bh - b * N_HEADS;

  const float* xr = x  + ((size_t)b * SEQ + t) * N_EMBD;
  const float* wq = Wq + (size_t)h * N_EMBD * HEAD_SIZE + d;
  const float* wk = Wk + (size_t)h * N_EMBD * HEAD_SIZE + d;
  const float* wv = Wv + (size_t)h * N_EMBD * HEAD_SIZE + d;

  float aq = 0.f, ak = 0.f, av = 0.f;
#pragma unroll 8
  for (int c = 0; c < N_EMBD; ++c) {
    float xv = xr[c];
    aq = fmaf(xv, wq[c * HEAD_SIZE], aq);
    ak = fmaf(xv, wk[c * HEAD_SIZE], ak);
    av = fmaf(xv, wv[c * HEAD_SIZE], av);
  }
  size_t row = ((size_t)bh * SEQ + t) * HEAD_SIZE + d;
  qo[row] = aq;
  ko[row] = ak;
  vT[((size_t)bh * HEAD_SIZE + d) * SEQ + t] = av;
}

// ---------------------------------------------------------------------------
// Kernel 2: fused causal flash attention. One wave32 per 16-query tile.
// Computes S^T = K x Q^T and O^T = V^T x P^T with V_WMMA_F32_16X16X4_F32.
// C/D layout: lane%16 = column (query), VGPR r = row (r + 8*(lane>=16)).
// A layout (16x4 f32): lane%16 = row, VGPR j: K = j + 2*(lane>=16).
// B layout (4x16 f32): lane%16 = col, VGPR j: K = j + 2*(lane>=16).
// ---------------------------------------------------------------------------
__global__ __launch_bounds__(32) void flash_attn_kernel(
    const float* __restrict__ q,
    const float* __restrict__ k,
    const float* __restrict__ vT,
    float* __restrict__ attn) {
  const int lane = threadIdx.x;      // 0..31
  const int col  = lane & 15;        // query column owned by this lane
  const int half = lane >> 4;        // 0: rows 0..7, 1: rows 8..15

  const int qb = blockIdx.x & (SEQ / 16 - 1);
  const int bh = blockIdx.x >> 8;    // SEQ/16 == 256
  const int t0 = qb * 16;

  const float* qbase = q  + (size_t)bh * SEQ * HEAD_SIZE;
  const float* kbase = k  + (size_t)bh * SEQ * HEAD_SIZE;
  const float* vbase = vT + (size_t)bh * HEAD_SIZE * SEQ;

  // Q^T B-operand fragments (loop invariant): Q^T[d][q] = Q[q][d].
  v2f qfrag[4];
#pragma unroll
  for (int c = 0; c < 4; ++c)
    qfrag[c] = *(const v2f*)(qbase + (size_t)(t0 + col) * HEAD_SIZE + 4 * c + 2 * half);

  v8f o = {};                         // O^T accumulators
  float m = -1e30f, l = 0.f;
  const float scale = 0.25f;          // 1/sqrt(16)

  for (int kb = 0; kb <= qb; ++kb) {
    const int kstart = kb * 16;

    // ---- S^T = K x Q^T : 4 WMMAs, contraction over d=16 ----
    v8f s = {};
#pragma unroll
    for (int c = 0; c < 4; ++c) {
      v2f af = *(const v2f*)(kbase + (size_t)(kstart + col) * HEAD_SIZE + 4 * c + 2 * half);
      s = __builtin_amdgcn_wmma_f32_16x16x4_f32(false, af, false, qfrag[c],
                                                (short)0, s, false, false);
    }

    float p8[8];
#pragma unroll
    for (int r = 0; r < 8; ++r) p8[r] = s[r] * scale;

    if (kb == qb) {                   // causal mask on the diagonal tile
#pragma unroll
      for (int r = 0; r < 8; ++r) {
        int key = kstart + r + 8 * half;
        if (key > t0 + col) p8[r] = -1e30f;
      }
    }

    // ---- online softmax (per-query = per-lane scalars) ----
    float mloc = p8[0];
#pragma unroll
    for (int r = 1; r < 8; ++r) mloc = fmaxf(mloc, p8[r]);
    mloc = fmaxf(mloc, __shfl_xor(mloc, 16, 32));
    float mnew  = fmaxf(m, mloc);
    float alpha = __expf(m - mnew);
    float sloc = 0.f;
#pragma unroll
    for (int r = 0; r < 8; ++r) { p8[r] = __expf(p8[r] - mnew); sloc += p8[r]; }
    l = l * alpha + sloc + __shfl_xor(sloc, 16, 32);
    m = mnew;
#pragma unroll
    for (int r = 0; r < 8; ++r) o[r] *= alpha;

    // ---- O^T += V^T x P^T : 4 WMMAs, contraction over 16 keys ----
#pragma unroll
    for (int c = 0; c < 4; ++c) {
      v2f af = *(const v2f*)(vbase + (size_t)col * SEQ + kstart + 4 * c + 2 * half);
      // Repack P^T (C-layout) into B-operand layout via cross-half shuffles.
      v2f bf;
      if (c < 2) {
        float u0 = __shfl_xor(p8[4 * c + 2], 16, 32);
        float u1 = __shfl_xor(p8[4 * c + 3], 16, 32);
        bf.x = half ? u0 : p8[4 * c + 0];
        bf.y = half ? u1 : p8[4 * c + 1];
      } else {
        float u0 = __shfl_xor(p8[4 * c - 8], 16, 32);
        float u1 = __shfl_xor(p8[4 * c - 7], 16, 32);
        bf.x = half ? p8[4 * c - 6] : u0;
        bf.y = half ? p8[4 * c - 5] : u1;
      }
      o = __builtin_amdgcn_wmma_f32_16x16x4_f32(false, af, false, bf,
                                                (short)0, o, false, false);
    }
  }

  // ---- normalize and store concat-head output ----
  float inv_l = 1.0f / l;
  int b = bh / N_HEADS;
  int h = bh - b * N_HEADS;
  float* outp = attn + ((size_t)(b * SEQ + t0 + col)) * N_EMBD + h * HEAD_SIZE + 8 * half;
#pragma unroll
  for (int r = 0; r < 8; ++r) outp[r] = o[r] * inv_l;
}

// ---------------------------------------------------------------------------
// Kernel 3: output projection  out = attn @ Wp^T + bp
// ---------------------------------------------------------------------------
__global__ void out_proj_kernel(const float* __restrict__ attn,
                                const float* __restrict__ Wp,
                                const float* __restrict__ bp,
                                float* __restrict__ out) {
  int idx = blockIdx.x * blockDim.x + threadIdx.x;   // B*T*96
  if (idx >= BATCH * SEQ * N_EMBD) return;
  int oc = idx % N_EMBD;
  int t  = idx / N_EMBD;
  const float* a = attn + (size_t)t * N_EMBD;
  const float* w = Wp + (size_t)oc * N_EMBD;
  float acc = bp[oc];
#pragma unroll 8
  for (int c = 0; c < N_EMBD; ++c) acc = fmaf(a[c], w[c], acc);
  out[idx] = acc;
}

// ---------------------------------------------------------------------------
extern "C" void kernel_launch(void* const* d_in, const int* in_sizes, int n_in,
                              void* d_out, int out_size, void* d_ws, size_t ws_size,
                              hipStream_t stream) {
  (void)in_sizes; (void)n_in; (void)out_size; (void)ws_size;
  const float* x  = (const float*)d_in[0];
  const float* Wq = (const float*)d_in[1];
  const float* Wk = (const float*)d_in[2];
  const float* Wv = (const float*)d_in[3];
  const float* Wp = (const float*)d_in[4];
  const float* bp = (const float*)d_in[5];

  float* ws = (float*)d_ws;
  const size_t S1 = (size_t)BATCH * N_HEADS * SEQ * HEAD_SIZE;  // 786432 floats
  float* qbuf = ws;
  float* kbuf = ws + S1;
  float* vTbf = ws + 2 * S1;
  float* abuf = ws + 3 * S1;   // attention output [B,T,96]

  {
    int threads = BATCH * N_HEADS * SEQ * HEAD_SIZE;
    qkv_proj_kernel<<<threads / 256, 256, 0, stream>>>(x, Wq, Wk, Wv, qbuf, kbuf, vTbf);
  }
  {
    int blocks = BATCH * N_HEADS * (SEQ / 16);     // 3072 waves
    flash_attn_kernel<<<blocks, 32, 0, stream>>>(qbuf, kbuf, vTbf, abuf);
  }
  {
    int n = BATCH * SEQ * N_EMBD;
    out_proj_kernel<<<(n + 255) / 256, 256, 0, stream>>>(abuf, Wp, bp, (float*)d_out);
  }
}